// WeightsLib2DDelta_64707977281755
// MI455X (gfx1250) — compile-verified
//
#include <hip/hip_runtime.h>
#include <hip/hip_bf16.h>
#include <math.h>

typedef __attribute__((ext_vector_type(2))) float v2f;
typedef __attribute__((ext_vector_type(8))) float v8f;

#define NPIX 262144  // 512*512

// ---------------------------------------------------------------------------
// K0: zero the reduction accumulators (sumsq1[32], dot[256], sumsq2[32])
// ---------------------------------------------------------------------------
__global__ void zero_kernel(float* __restrict__ p, int n) {
    int i = blockIdx.x * blockDim.x + threadIdx.x;
    if (i < n) p[i] = 0.f;
}

// ---------------------------------------------------------------------------
// K1: xt = x @ ctx_w^T + ctx_b via V_WMMA_F32_16X16X4_F32 (M=16 pad of B=4,
// N=14 tiles of 16 covering 210+pad, K=512 in 128 chained WMMAs), then the
// tanh gate, the per-(b,r,comp) modulation params, and the softmax weights.
// One block, 14 waves (448 threads).
// ---------------------------------------------------------------------------
__global__ __launch_bounds__(448) void param_kernel(
    const float* __restrict__ x, const float* __restrict__ ctx_w,
    const float* __restrict__ ctx_b, float* __restrict__ params,
    float* __restrict__ wr_out)
{
    __shared__ float s_xt[4 * 210];
    __shared__ float s_g[4 * 105];
    const int tid = threadIdx.x, wv = tid >> 5, lane = tid & 31;
    const int m = lane & 15, koff = (lane >> 4) << 1;

    // A fragment: row m of x (rows >=4 are zero padding)
    const float* arow = x + (size_t)(m < 4 ? m : 0) * 512;
    const float  asc  = (m < 4) ? 1.f : 0.f;
    // B fragment: column n = wv*16 + m of ctx_w^T == row n of ctx_w
    const int    ncol = wv * 16 + m;
    const float* brow = ctx_w + (size_t)(ncol < 210 ? ncol : 0) * 512;
    const float  bsc  = (ncol < 210) ? 1.f : 0.f;

    v8f c = {};
    for (int k0 = 0; k0 < 512; k0 += 4) {
        float2 ta = *(const float2*)(arow + k0 + koff);
        float2 tb = *(const float2*)(brow + k0 + koff);
        v2f a;  a.x  = ta.x * asc; a.y  = ta.y * asc;
        v2f bb; bb.x = tb.x * bsc; bb.y = tb.y * bsc;
        c = __builtin_amdgcn_wmma_f32_16x16x4_f32(false, a, false, bb,
                                                  (short)0, c, false, false);
    }
    // C layout: VGPR v, lanes 0-15 -> row M=v, col = lane. Rows 0..3 are real.
    if (lane < 16) {
#pragma unroll
        for (int v = 0; v < 4; ++v) {
            int nn = wv * 16 + lane;
            if (nn < 210) s_xt[v * 210 + nn] = c[v] + ctx_b[nn];
        }
    }
    __syncthreads();
    // gate: xt[:, :105] * tanh(xt[:, 105:210])
    if (tid < 420) {
        int b = tid / 105, j = tid % 105;
        s_g[b * 105 + j] = s_xt[b * 210 + j] * tanhf(s_xt[b * 210 + 105 + j]);
    }
    __syncthreads();
    // modulation params: per (b, r, comp u): cR,cI,biasR,biasI,sR,sI
    if (tid < 64) {
        int b = tid >> 4, r = (tid >> 1) & 7, u = tid & 1;
        const float* g = s_g + b * 105 + r * 12 + u * 2;
        float* P = params + (size_t)(b * 8 + r) * 12 + u * 6;
        P[0] = g[0] + 0.5f;  // c.re + shift_lerp
        P[1] = g[1];         // c.im
        P[2] = g[4];         // bias.re
        P[3] = g[5];         // bias.im
        P[4] = g[8] + 1.f;   // scale.re + shift_scale
        P[5] = g[9];         // scale.im
    } else if (tid < 68) {
        // softmax weights wr: l2norm(xtg[96:105] with last += 1/sqrt(8)), softmax
        int b = tid - 64;
        float v[9], ss = 0.f;
#pragma unroll
        for (int j = 0; j < 9; ++j) v[j] = s_g[b * 105 + 96 + j];
        v[8] += 0.35355339059327373f;
#pragma unroll
        for (int j = 0; j < 9; ++j) ss += v[j] * v[j];
        float inv = 1.f / fmaxf(sqrtf(ss), 1e-12f);
        float mx = -3.4e38f;
#pragma unroll
        for (int j = 0; j < 9; ++j) { v[j] *= inv; mx = fmaxf(mx, v[j]); }
        float se = 0.f;
#pragma unroll
        for (int j = 0; j < 9; ++j) { v[j] = expf(v[j] - mx); se += v[j]; }
        float is = 1.f / se;
#pragma unroll
        for (int j = 0; j < 9; ++j) wr_out[b * 9 + j] = v[j] * is;
    }
}

// ---------------------------------------------------------------------------
// K2: stream the 67MB weights tensor exactly ONCE; each thread's 4 float2
// loads feed all 4 batch elements. Produces wf[b][r][p] and sum(wf^2) per
// (b,r) via LDS + global float atomics.  grid (512, r=8), 256 threads.
// ---------------------------------------------------------------------------
__global__ __launch_bounds__(256) void wf_kernel(
    const float* __restrict__ W, const float* __restrict__ params,
    float* __restrict__ wf, float* __restrict__ sumsq1)
{
    const int r = blockIdx.y;
    const int tid = threadIdx.x;
    __shared__ float s_par[48];  // [b][u][6]
    __shared__ float s_acc[4];
    if (tid < 48) { int b = tid / 12, k = tid % 12; s_par[tid] = params[(size_t)(b * 8 + r) * 12 + k]; }
    if (tid < 4) s_acc[tid] = 0.f;
    __syncthreads();

    float acc[4] = {0.f, 0.f, 0.f, 0.f};
    const size_t rbase = (size_t)r * 2097152u;  // r stride in elements
    for (int p = blockIdx.x * 256 + tid; p < NPIX; p += 512 * 256) {
        const size_t base = rbase + (size_t)p * 2u;
        __builtin_prefetch(W + base + (size_t)(512 * 256) * 2u, 0, 0);
        float2 w00 = *(const float2*)(W + base);             // j=0, comp=0
        float2 w01 = *(const float2*)(W + base + 524288u);   // j=0, comp=1
        float2 w10 = *(const float2*)(W + base + 1048576u);  // j=1, comp=0
        float2 w11 = *(const float2*)(W + base + 1572864u);  // j=1, comp=1
        const float dr0 = w10.x - w00.x, di0 = w10.y - w00.y;
        const float dr1 = w11.x - w01.x, di1 = w11.y - w01.y;
#pragma unroll
        for (int b = 0; b < 4; ++b) {
            const float* P = &s_par[b * 12];
            float lr0 = w00.x + P[0] * dr0 - P[1] * di0;
            float li0 = w00.y + P[0] * di0 + P[1] * dr0;
            float zr0 = P[2] + lr0 * P[4] - li0 * P[5];
            float zi0 = P[3] + lr0 * P[5] + li0 * P[4];
            float lr1 = w01.x + P[6] * dr1 - P[7] * di1;
            float li1 = w01.y + P[6] * di1 + P[7] * dr1;
            float zr1 = P[8] + lr1 * P[10] - li1 * P[11];
            float zi1 = P[9] + lr1 * P[11] + li1 * P[10];
            // w = <num, den/|den|>
            float w = (zr0 * zr1 + zi0 * zi1) *
                      rsqrtf(fmaxf(zr1 * zr1 + zi1 * zi1, 1e-12f));
            if (fabsf(w) < 1e-12f) w = copysignf(1e-12f, w);  // _stable (fwd)
            wf[(size_t)(b * 8 + r) * NPIX + p] = w;
            acc[b] += w * w;
        }
    }
#pragma unroll
    for (int b = 0; b < 4; ++b) atomicAdd(&s_acc[b], acc[b]);
    __syncthreads();
    if (tid < 4) atomicAdd(&sumsq1[tid * 8 + r], s_acc[tid]);
}

// ---------------------------------------------------------------------------
// K3: Gram matrix sim = wf_n . wf_n^T  (M=8,N=8,K=262144) as a chained
// V_WMMA_F32_16X16X4_F32 reduction. A-fragment == B-fragment (symmetric).
// grid (32, b=4), 8 waves/block, 256 chained WMMAs per wave.
// ---------------------------------------------------------------------------
__global__ __launch_bounds__(256) void sim_kernel(
    const float* __restrict__ wf, const float* __restrict__ sumsq1,
    float* __restrict__ dotm)
{
    const int b = blockIdx.y;
    const int lane = threadIdx.x & 31;
    const int gw = blockIdx.x * 8 + (threadIdx.x >> 5);  // 0..255
    const int m = lane & 15, koff = (lane >> 4) << 1;
    const int mm = (m < 8) ? m : 0;
    const float sq = sumsq1[b * 8 + mm];
    const float sc = (m < 8) ? (1.f / fmaxf(sqrtf(sq), 1e-12f)) : 0.f;
    const float* row = wf + (size_t)(b * 8 + mm) * NPIX;

    v8f c = {};
    for (int it = 0; it < 256; ++it) {
        const int k0 = ((it << 8) + gw) << 2;  // K-chunk of 4
        float2 t = *(const float2*)(row + k0 + koff);
        v2f a; a.x = t.x * sc; a.y = t.y * sc;
        c = __builtin_amdgcn_wmma_f32_16x16x4_f32(false, a, false, a,
                                                  (short)0, c, false, false);
    }
    // C: VGPR v, lanes 0-15 -> row v, col lane. Keep the real 8x8 corner.
    if (lane < 8) {
#pragma unroll
        for (int v = 0; v < 8; ++v)
            atomicAdd(&dotm[(size_t)(b * 8 + v) * 8 + lane], c[v]);
    }
}

// ---------------------------------------------------------------------------
// K4: repulsion + renorm prep, in place:
//   wf2 = stable( wf_n - (sim_offdiag . wf_n) * REPULSION/(sqrt(r)*(r-1)) )
// plus sum(wf2^2) per (b,r).  grid (256, b=4), 256 threads.
// ---------------------------------------------------------------------------
__global__ __launch_bounds__(256) void rep_kernel(
    float* __restrict__ wf, const float* __restrict__ sumsq1,
    const float* __restrict__ dotm, float* __restrict__ sumsq2)
{
    const int b = blockIdx.y;
    const int tid = threadIdx.x;
    __shared__ float s_sim[64], s_inv[8], s_acc[8];
    if (tid < 8) {
        s_inv[tid] = 1.f / fmaxf(sqrtf(sumsq1[b * 8 + tid]), 1e-12f);
        s_acc[tid] = 0.f;
    }
    if (tid < 64) {
        int r = tid >> 3, s = tid & 7;
        float nr = fmaxf(sqrtf(sumsq1[b * 8 + r]), 1e-12f);
        float ns = fmaxf(sqrtf(sumsq1[b * 8 + s]), 1e-12f);
        s_sim[tid] = (r == s) ? 0.f : dotm[(size_t)(b * 8 + r) * 8 + s] / (nr * ns);
    }
    __syncthreads();

    const float COEF = 0.1f / (7.f * 2.8284271247461903f);  // REPULSION/((r-1)*sqrt(r))
    float acc[8] = {0.f, 0.f, 0.f, 0.f, 0.f, 0.f, 0.f, 0.f};
    for (int p = blockIdx.x * 256 + tid; p < NPIX; p += 256 * 256) {
        float v[8];
#pragma unroll
        for (int s = 0; s < 8; ++s)
            v[s] = wf[(size_t)(b * 8 + s) * NPIX + p] * s_inv[s];
#pragma unroll
        for (int r = 0; r < 8; ++r) {
            float rp = 0.f;
#pragma unroll
            for (int s = 0; s < 8; ++s) rp += s_sim[r * 8 + s] * v[s];
            float o = v[r] - rp * COEF;
            if (fabsf(o) < 1e-12f) o = copysignf(1e-12f, o);
            wf[(size_t)(b * 8 + r) * NPIX + p] = o;  // safe: this thread owns pixel p
            acc[r] += o * o;
        }
    }
#pragma unroll
    for (int r = 0; r < 8; ++r) atomicAdd(&s_acc[r], acc[r]);
    __syncthreads();
    if (tid < 8) atomicAdd(&sumsq2[b * 8 + tid], s_acc[tid]);
}

// ---------------------------------------------------------------------------
// K5: out[b][p] = sum_r wf2_n[b][r][p] * wr[b][r]   (rank-8 zero channel drops)
// ---------------------------------------------------------------------------
__global__ __launch_bounds__(256) void out_kernel(
    const float* __restrict__ wf, const float* __restrict__ sumsq2,
    const float* __restrict__ wr, float* __restrict__ out)
{
    const int b = blockIdx.y;
    const int tid = threadIdx.x;
    __shared__ float s_c[8];
    if (tid < 8)
        s_c[tid] = wr[b * 9 + tid] / fmaxf(sqrtf(sumsq2[b * 8 + tid]), 1e-12f);
    __syncthreads();
    for (int p = blockIdx.x * 256 + tid; p < NPIX; p += 256 * 256) {
        float o = 0.f;
#pragma unroll
        for (int r = 0; r < 8; ++r)
            o += wf[(size_t)(b * 8 + r) * NPIX + p] * s_c[r];
        out[(size_t)b * NPIX + p] = o;
    }
}

// ---------------------------------------------------------------------------
extern "C" void kernel_launch(void* const* d_in, const int* in_sizes, int n_in,
                              void* d_out, int out_size, void* d_ws, size_t ws_size,
                              hipStream_t stream)
{
    const float* x     = (const float*)d_in[0];  // (4, 512)
    const float* ctx_w = (const float*)d_in[1];  // (210, 512)
    const float* ctx_b = (const float*)d_in[2];  // (210,)
    const float* W     = (const float*)d_in[3];  // (1,8,2,2,512,512,2)

    float* ws     = (float*)d_ws;
    float* params = ws;          // 384: [b][r][u][6]
    float* wr     = ws + 384;    // 36:  [b][9]
    float* sumsq1 = ws + 420;    // 32
    float* dotm   = ws + 452;    // 256: [b][r][s]
    float* sumsq2 = ws + 708;    // 32
    float* wf     = ws + 1024;   // 8,388,608: [b][r][NPIX]  (~33.6 MB)

    zero_kernel <<<1, 320, 0, stream>>>(sumsq1, 320);
    param_kernel<<<1, 448, 0, stream>>>(x, ctx_w, ctx_b, params, wr);
    wf_kernel   <<<dim3(512, 8), 256, 0, stream>>>(W, params, wf, sumsq1);
    sim_kernel  <<<dim3(32, 4),  256, 0, stream>>>(wf, sumsq1, dotm);
    rep_kernel  <<<dim3(256, 4), 256, 0, stream>>>(wf, sumsq1, dotm, sumsq2);
    out_kernel  <<<dim3(256, 4), 256, 0, stream>>>(wf, sumsq2, wr, (float*)d_out);
}